// RoIAlignAvg_18073222382194
// MI455X (gfx1250) — compile-verified
//
#include <hip/hip_runtime.h>

// ---------------------------------------------------------------------------
// RoIAlign3D (5x8x8 grid, trilinear) + fused avg_pool3d(k=2,s=1)
// MI455X / gfx1250, wave32.
//
// Roofline: 103 MB output stores (HBM), ~1.3 GB gathered feature reads that
// stay resident in the 192 MB L2 (features = 268 MB). Memory-bound; FLOPs
// negligible. Spatial (y,x) pooling runs as a per-16-channel-tile matmul on
// V_WMMA_F32_16X16X4_F32 (exact f32 math); temporal pooling in VALU.
// B-matrix (pool weights) is generated branchlessly from a per-lane 64-bit
// mask and hoisted out of the t loop (kb-outer / t-inner interchange).
// Reuse hints are OFF: adjacent WMMAs are never identical instructions, so
// the ISA's matrix-reuse legality condition cannot be met here.
// ---------------------------------------------------------------------------

typedef __attribute__((ext_vector_type(2))) float v2f;
typedef __attribute__((ext_vector_type(8))) float v8f;

#define NROIS 256
#define CH    512
#define TDIM  8
#define HDIM  64
#define WDIM  64
#define NT    5          // time grid points
#define NHW   64         // 8*8 spatial grid points
#define CTILE 16
#define KSTRIDE 17       // padded channel stride in LDS (bank-conflict-free)
#define PLANE (NHW*KSTRIDE + 16)  // odd-ish plane stride (1104 floats)
#define SPATIAL_SCALE 0.0625f

// -------------------- Kernel 1: per-ROI sampling tables ---------------------
// Layout per ROI (64 dwords): [0]=bidx, [1..5]=tlo, [6..10]=thi, [11..15]=tf,
// [16..23]=ylo, [24..31]=yhi, [32..39]=yf, [40..47]=xlo, [48..55]=xhi, [56..63]=xf
__global__ __launch_bounds__(256) void roi_tables_kernel(
    const float* __restrict__ rois, int* __restrict__ tbl) {
  int n = threadIdx.x;
  if (n >= NROIS) return;
  const float* r = rois + n * 7;
  int*   o  = tbl + n * 64;
  float* of = reinterpret_cast<float*>(o);

  o[0] = (int)r[0];
  float x1 = r[1] * SPATIAL_SCALE, y1 = r[2] * SPATIAL_SCALE, t1 = r[3];
  float x2 = r[4] * SPATIAL_SCALE, y2 = r[5] * SPATIAL_SCALE, t2 = r[6];

  float st = (t2 - t1) * (1.0f / 4.0f);
  #pragma unroll
  for (int i = 0; i < 5; ++i) {
    float c = fminf(fmaxf(t1 + st * (float)i, 0.0f), (float)(TDIM - 1));
    int lo = (int)floorf(c);
    int hi = min(lo + 1, TDIM - 1);
    o[1 + i] = lo; o[6 + i] = hi; of[11 + i] = c - (float)lo;
  }
  float sy = (y2 - y1) * (1.0f / 7.0f);
  #pragma unroll
  for (int j = 0; j < 8; ++j) {
    float c = fminf(fmaxf(y1 + sy * (float)j, 0.0f), (float)(HDIM - 1));
    int lo = (int)floorf(c);
    int hi = min(lo + 1, HDIM - 1);
    o[16 + j] = lo; o[24 + j] = hi; of[32 + j] = c - (float)lo;
  }
  float sx = (x2 - x1) * (1.0f / 7.0f);
  #pragma unroll
  for (int j = 0; j < 8; ++j) {
    float c = fminf(fmaxf(x1 + sx * (float)j, 0.0f), (float)(WDIM - 1));
    int lo = (int)floorf(c);
    int hi = min(lo + 1, WDIM - 1);
    o[40 + j] = lo; o[48 + j] = hi; of[56 + j] = c - (float)lo;
  }
}

// ---------------- Kernel 2: fused sample + WMMA pool ------------------------
// Block = (roi, 16-channel tile); 128 threads = 4 waves.
__global__ __launch_bounds__(128) void roi_align_pool_kernel(
    const float* __restrict__ feat, const int* __restrict__ tbl,
    float* __restrict__ out) {
  __shared__ int   lt[64];
  __shared__ float s[NT * PLANE];   // grid samples: s[t*PLANE + k*17 + c]

  const int n   = blockIdx.x >> 5;          // roi
  const int c0  = (blockIdx.x & 31) * CTILE;
  const int tid = threadIdx.x;

  if (tid < 64) lt[tid] = tbl[n * 64 + tid];
  __syncthreads();
  const float* ltf = reinterpret_cast<const float*>(lt);
  const int b = lt[0];
  const float* fb = feat + (size_t)b * CH * TDIM * HDIM * WDIM;

  // ---- Stage 1: trilinear samples -> LDS. 640 rows (16c x 5t x 8y), each
  // thread handles 5 rows; inner loop walks the 8 x-grid points of a row so
  // the 8 gathers per point hit nearby columns of the same feature rows.
  #pragma unroll
  for (int i = 0; i < 5; ++i) {
    const int r   = tid + i * 128;
    const int c   = r / 40;
    const int rem = r - c * 40;
    const int t   = rem >> 3;
    const int y   = rem & 7;

    const int   tlo = lt[1 + t],  thi = lt[6 + t];
    const float tf  = ltf[11 + t];
    const int   ylo = lt[16 + y], yhi = lt[24 + y];
    const float yf  = ltf[32 + y];
    const float w00 = (1.0f - tf) * (1.0f - yf);
    const float w01 = (1.0f - tf) * yf;
    const float w10 = tf * (1.0f - yf);
    const float w11 = tf * yf;

    const float* fc  = fb + (size_t)(c0 + c) * (TDIM * HDIM * WDIM);
    const float* p00 = fc + (tlo * HDIM + ylo) * WDIM;
    const float* p01 = fc + (tlo * HDIM + yhi) * WDIM;
    const float* p10 = fc + (thi * HDIM + ylo) * WDIM;
    const float* p11 = fc + (thi * HDIM + yhi) * WDIM;

    float* srow = s + t * PLANE + (y * 8) * KSTRIDE + c;
    #pragma unroll
    for (int x = 0; x < 8; ++x) {
      const int   xlo = lt[40 + x], xhi = lt[48 + x];
      const float xf  = ltf[56 + x];
      const float wx0 = 1.0f - xf;
      float v = w00 * (p00[xlo] * wx0 + p00[xhi] * xf)
              + w01 * (p01[xlo] * wx0 + p01[xhi] * xf)
              + w10 * (p10[xlo] * wx0 + p10[xhi] * xf)
              + w11 * (p11[xlo] * wx0 + p11[xhi] * xf);
      srow[x * KSTRIDE] = v;
    }
  }
  __syncthreads();

  // ---- Stage 2: spatial avg-pool as WMMA f32 16x16x4 matmul.
  // out_yx[c,q] = sum_k s[c,k] * P[k,q], P[(ky,kx),(qy,qx)] = 0.25 when
  // ky in {qy,qy+1} and kx in {qx,qx+1}; q in [0,49). Each wave owns one
  // 16-wide N tile of q. A layout (16x4 f32): lane<16 -> M=lane, K={0,1};
  // lane>=16 -> M=lane-16, K={2,3}. B (4x16) mirrored on K/N.
  const int wave = tid >> 5;
  const int lane = tid & 31;
  const int c    = lane & 15;
  const int half = lane >> 4;
  const int q    = wave * 16 + (lane & 15);
  const int qy   = q / 7;
  const int qx   = q - qy * 7;

  // Per-lane 64-bit nonzero-weight mask over k = 8*ky + kx: rows qy and qy+1,
  // columns qx and qx+1 (branchless; zero when this lane's q >= 49).
  unsigned long long wmask = 0ull;
  {
    const unsigned long long row = (unsigned long long)(3u << qx) & 0xffull;
    unsigned long long m = (row << (8 * qy)) | (row << (8 * qy + 8));
    wmask = (q < 49) ? m : 0ull;
  }

  v8f g[NT] = {v8f{}, v8f{}, v8f{}, v8f{}, v8f{}};
  const int koff = half * 2;  // this half-wave's K sub-pair within each K=4 chunk
  #pragma unroll
  for (int kb = 0; kb < 16; ++kb) {
    const int k0 = kb * 4 + koff;
    const unsigned two = (unsigned)(wmask >> k0) & 3u;   // bits for k0, k0+1
    v2f bb;
    bb.x = (two & 1u) ? 0.25f : 0.0f;
    bb.y = (two & 2u) ? 0.25f : 0.0f;
    #pragma unroll
    for (int t = 0; t < NT; ++t) {
      const float* sp = s + t * PLANE;
      v2f a;
      a.x = sp[(k0    ) * KSTRIDE + c];
      a.y = sp[(k0 + 1) * KSTRIDE + c];
      g[t] = __builtin_amdgcn_wmma_f32_16x16x4_f32(
          /*neg_a=*/false, a, /*neg_b=*/false, bb,
          /*c_mod=*/(short)0, g[t], /*reuse_a=*/false, /*reuse_b=*/false);
    }
  }

  // ---- Temporal pooling: out[t'] = 0.5*(g[t'] + g[t'+1]), t' in [0,4).
  v8f oacc[4];
  #pragma unroll
  for (int tp = 0; tp < 4; ++tp) oacc[tp] = (g[tp] + g[tp + 1]) * 0.5f;

  // ---- Store: D layout 16x16 f32: VGPR v -> M = v + 8*half, N = lane&15.
  if (q < 49) {
    #pragma unroll
    for (int tp = 0; tp < 4; ++tp) {
      #pragma unroll
      for (int v = 0; v < 8; ++v) {
        const int M = v + half * 8;
        const size_t idx = ((size_t)n * CH + (size_t)(c0 + M)) * 196
                         + (size_t)(tp * 49 + q);
        __builtin_nontemporal_store(oacc[tp][v], out + idx);
      }
    }
  }
}

// ---------------------------------------------------------------------------
extern "C" void kernel_launch(void* const* d_in, const int* in_sizes, int n_in,
                              void* d_out, int out_size, void* d_ws, size_t ws_size,
                              hipStream_t stream) {
  const float* feat = (const float*)d_in[0];   // [4,512,8,64,64] f32
  const float* rois = (const float*)d_in[1];   // [256,7] f32
  float* out = (float*)d_out;                  // [256,512,4,7,7] f32
  int*   tbl = (int*)d_ws;                     // 256 * 64 dwords = 64 KB

  roi_tables_kernel<<<dim3(1), dim3(256), 0, stream>>>(rois, tbl);
  roi_align_pool_kernel<<<dim3(NROIS * (CH / CTILE)), dim3(128), 0, stream>>>(
      feat, tbl, out);
}